// PointNetFeaturePropagationQ_69982197121139
// MI455X (gfx1250) — compile-verified
//
#include <hip/hip_runtime.h>

typedef __attribute__((ext_vector_type(2))) float v2f;
typedef __attribute__((ext_vector_type(8))) float v8f;

#define BATCH 8
#define NPTS  4096
#define SPTS  1024
#define D1C   128
#define D2C   256
#define CIN   384
#define COUT  256
#define QEPS  1e-5f
// mean over (3B, N) = 24*4096 elements per channel
#define INV_CNT (1.0f / 98304.0f)

// ---------------------------------------------------------------- zero
__global__ void zero_kernel(float* __restrict__ p, int n) {
  int t = blockIdx.x * blockDim.x + threadIdx.x;
  if (t < n) p[t] = 0.f;
}

// ---------------------------------------------------------------- 3-NN + weights
// grid: BATCH * (NPTS/256) blocks, 256 threads
__global__ __launch_bounds__(256)
void knn_kernel(const float* __restrict__ xyz1, const float* __restrict__ xyz2,
                int* __restrict__ idx3, float* __restrict__ w3) {
  __shared__ float sx[SPTS], sy[SPTS], sz[SPTS], sn[SPTS];
  const int b = blockIdx.x >> 4;                       // 16 blocks per batch
  const int n = ((blockIdx.x & 15) << 8) + threadIdx.x;
  for (int s = threadIdx.x; s < SPTS; s += 256) {
    float xs = xyz2[(b * 3 + 0) * SPTS + s];
    float ys = xyz2[(b * 3 + 1) * SPTS + s];
    float zs = xyz2[(b * 3 + 2) * SPTS + s];
    sx[s] = xs; sy[s] = ys; sz[s] = zs;
    sn[s] = xs * xs + ys * ys + zs * zs;
  }
  __syncthreads();
  const float x = xyz1[(b * 3 + 0) * NPTS + n];
  const float y = xyz1[(b * 3 + 1) * NPTS + n];
  const float z = xyz1[(b * 3 + 2) * NPTS + n];
  const float xn = x * x + y * y + z * z;
  float d0 = 3.4e38f, d1 = 3.4e38f, d2 = 3.4e38f;
  int   i0 = 0, i1 = 0, i2 = 0;
  for (int s = 0; s < SPTS; ++s) {
    // same expansion as the reference: |src|^2 + |dst|^2 - 2 src.dst
    float d = xn + sn[s] - 2.f * (x * sx[s] + y * sy[s] + z * sz[s]);
    if (d < d2) {
      if (d < d1) {
        if (d < d0) { d2 = d1; i2 = i1; d1 = d0; i1 = i0; d0 = d; i0 = s; }
        else        { d2 = d1; i2 = i1; d1 = d;  i1 = s; }
      } else        { d2 = d;  i2 = s; }
    }
  }
  d0 = fmaxf(d0, 1e-10f); d1 = fmaxf(d1, 1e-10f); d2 = fmaxf(d2, 1e-10f);
  float w0 = 1.f / d0, w1 = 1.f / d1, w2 = 1.f / d2;
  float inv = 1.f / (w0 + w1 + w2);
  const int base = (b * NPTS + n) * 3;
  idx3[base + 0] = i0; idx3[base + 1] = i1; idx3[base + 2] = i2;
  w3[base + 0] = w0 * inv; w3[base + 1] = w1 * inv; w3[base + 2] = w2 * inv;
}

// ---------------------------------------------------------------- GEMM
// out[bq][o][n] = sum_c W[o][c] * X[bq][c][n]
//   FUSE_INTERP: X = concat(points1 rows 0..127, interp(points2) rows 128..383)
//   else:        X = bsrc (activated hidden)
// LDS B-tile kept TRANSPOSED [n][k] (stride 34) so each WMMA B-fragment is one
// aligned b64 LDS read (fuses to ds_load_2addr_b64, no repack movs).
// Double-buffered; staged with per-lane global_load_async_to_lds_b32
// (ASYNCcnt, lane-scatter on the LDS side, coalesced on the global side);
// interpolated channels staged via VALU gather + ds_store.
// grid: 24 * 64 blocks (one per (bq, 64-point n-block)), 256 threads = 8 waves
template <int KDIM, bool FUSE_INTERP>
__global__ __launch_bounds__(256)
void gemm_kernel(const float* __restrict__ W,
                 const float* __restrict__ bsrc,
                 const float* __restrict__ points2,
                 const int*   __restrict__ idx3,
                 const float* __restrict__ w3,
                 float* __restrict__ out,
                 float* __restrict__ m2sum) {
  constexpr int SRC_ROWS = FUSE_INTERP ? D1C : KDIM;
  constexpr int LDB = 34;              // padded K-stride (even: keeps b64 aligned)
  constexpr int CHUNKS = KDIM / 32;
  __shared__ float Bs[2][64 * LDB];    // [buffer][n][k], transposed
  __shared__ float sqs[COUT];          // per-channel sum of squares

  const int tid  = threadIdx.x;
  const int bq   = blockIdx.x >> 6;
  const int n0g  = (blockIdx.x & 63) << 6;
  const int wave = tid >> 5;
  const int lane = tid & 31;
  const int half = lane >> 4;
  const int lr   = lane & 15;
  // staging mapping: 64 n x 4 k-rows per pass
  const int nl   = tid & 63;
  const int kr   = tid >> 6;
  const int ng   = n0g + nl;

  sqs[tid] = 0.f;

  int   gi0 = 0, gi1 = 0, gi2 = 0;
  float gw0 = 0.f, gw1 = 0.f, gw2 = 0.f;
  if constexpr (FUSE_INTERP) {
    const int b    = bq / 3;
    const int base = (b * NPTS + ng) * 3;
    gi0 = idx3[base + 0]; gi1 = idx3[base + 1]; gi2 = idx3[base + 2];
    gw0 = w3[base + 0];   gw1 = w3[base + 1];   gw2 = w3[base + 2];
  }

  v8f acc[2][4] = {};

  // async DMA stage of one 32x64 chunk: per-lane b32, global side coalesced
  // (consecutive lanes -> consecutive n), LDS side scatters into [n][k]
  auto stage_async = [&](int kc, int nb) {
#pragma unroll
    for (int p = 0; p < 8; ++p) {
      const float* g = bsrc + (bq * SRC_ROWS + kc + p * 4 + kr) * NPTS + ng;
      unsigned lds = (unsigned)(unsigned long long)&Bs[nb][nl * LDB + p * 4 + kr];
      asm volatile("global_load_async_to_lds_b32 %0, %1, off"
                   :: "v"(lds), "v"(g) : "memory");
    }
  };
  // VALU gather + interpolate stage (channels >= 128 of layer 1)
  auto stage_interp = [&](int kc, int nb) {
#pragma unroll
    for (int p = 0; p < 8; ++p) {
      const int c = kc + p * 4 + kr;
      const float* r = points2 + (bq * D2C + (c - D1C)) * SPTS;
      Bs[nb][nl * LDB + p * 4 + kr] = gw0 * r[gi0] + gw1 * r[gi1] + gw2 * r[gi2];
    }
  };

  // prologue: chunk 0 is always a plain copy (c < 128)
  stage_async(0, 0);

  int cur = 0;
#pragma unroll 1
  for (int ch = 0; ch < CHUNKS; ++ch) {
    // our own async loads must land before the barrier makes them visible
    asm volatile("s_wait_asynccnt 0x0" ::: "memory");
    __syncthreads();
    // prefetch next chunk into the other buffer while we compute this one
    if (ch + 1 < CHUNKS) {
      const int kc = (ch + 1) * 32;
      if (FUSE_INTERP && kc >= D1C) stage_interp(kc, cur ^ 1);
      else                          stage_async(kc, cur ^ 1);
    }
    // ---- 64 WMMAs per wave on buffer `cur` ----
    const int kc0 = ch * 32;
#pragma unroll
    for (int mt = 0; mt < 2; ++mt) {
      const int arow = wave * 32 + mt * 16 + lr;
      const float* wrow = W + arow * KDIM + kc0 + 2 * half;
#pragma unroll
      for (int ks = 0; ks < 32; ks += 4) {
        // A 16x4: lane = M row, half selects K pair {0,1}/{2,3}
        v2f a = *(const v2f*)(wrow + ks);
#pragma unroll
        for (int nt = 0; nt < 4; ++nt) {
          // B 4x16: lane = N col, half selects K pair; LDS holds B transposed
          v2f bf = *(const v2f*)(&Bs[cur][(nt * 16 + lr) * LDB + ks + 2 * half]);
          acc[mt][nt] = __builtin_amdgcn_wmma_f32_16x16x4_f32(
              false, a, false, bf, (short)0, acc[mt][nt], false, false);
        }
      }
    }
    cur ^= 1;
  }

  // ---- epilogue: store D, accumulate per-channel sum(x^2) ----
#pragma unroll
  for (int mt = 0; mt < 2; ++mt) {
    const int chbase = wave * 32 + mt * 16 + 8 * half;   // C/D row = r + 8*half
    float ps[8];
#pragma unroll
    for (int r = 0; r < 8; ++r) ps[r] = 0.f;
#pragma unroll
    for (int nt = 0; nt < 4; ++nt)
#pragma unroll
      for (int r = 0; r < 8; ++r) { float v = acc[mt][nt][r]; ps[r] += v * v; }
#pragma unroll
    for (int r = 0; r < 8; ++r) atomicAdd(&sqs[chbase + r], ps[r]);
#pragma unroll
    for (int nt = 0; nt < 4; ++nt) {
      const int nc = n0g + nt * 16 + lr;
#pragma unroll
      for (int r = 0; r < 8; ++r)
        out[(bq * COUT + chbase + r) * NPTS + nc] = acc[mt][nt][r];
    }
  }
  __syncthreads();
  atomicAdd(&m2sum[tid], sqs[tid]);
}

// ---------------------------------------------------------------- QBN (RMS) + quaternion ReLU, in place
// grid: BATCH*COUT*NPTS/256 blocks
__global__ __launch_bounds__(256)
void qbn_qrelu_kernel(float* __restrict__ x, const float* __restrict__ m2sum) {
  const int t = blockIdx.x * 256 + threadIdx.x;
  const int n = t & (NPTS - 1);
  const int c = (t >> 12) & (COUT - 1);
  const int b = t >> 20;
  const float inv = rsqrtf(m2sum[c] * INV_CNT + QEPS);
  const int i0 = ((b * 3 + 0) * COUT + c) * NPTS + n;
  const int i1 = ((b * 3 + 1) * COUT + c) * NPTS + n;
  const int i2 = ((b * 3 + 2) * COUT + c) * NPTS + n;
  float v0 = x[i0] * inv;
  float v1 = x[i1] * inv;
  float v2 = x[i2] * inv;
  float mod   = sqrtf(v0 * v0 + v1 * v1 + v2 * v2);
  float coeff = mod / fmaxf(mod, 1.f);
  x[i0] = v0 * coeff;
  x[i1] = v1 * coeff;
  x[i2] = v2 * coeff;
}

// ---------------------------------------------------------------- launcher
extern "C" void kernel_launch(void* const* d_in, const int* in_sizes, int n_in,
                              void* d_out, int out_size, void* d_ws, size_t ws_size,
                              hipStream_t stream) {
  const float* xyz1    = (const float*)d_in[0];
  const float* xyz2    = (const float*)d_in[1];
  const float* points1 = (const float*)d_in[2];
  const float* points2 = (const float*)d_in[3];
  const float* W0      = (const float*)d_in[4];
  const float* W1      = (const float*)d_in[5];
  float* out = (float*)d_out;

  char* ws   = (char*)d_ws;
  int*   idx3 = (int*)(ws);                    // 8*4096*3 ints   = 384 KiB
  float* w3   = (float*)(ws + 393216);         // 8*4096*3 floats = 384 KiB
  float* m2a  = (float*)(ws + 786432);         // 256 floats
  float* m2b  = (float*)(ws + 787456);         // 256 floats
  float* H1   = (float*)(ws + 788480);         // 24*256*4096 floats = 96 MiB

  // zero the two per-channel accumulators (ws is poisoned, not re-zeroed)
  zero_kernel<<<1, 512, 0, stream>>>(m2a, 512);

  knn_kernel<<<BATCH * (NPTS / 256), 256, 0, stream>>>(xyz1, xyz2, idx3, w3);

  gemm_kernel<CIN, true><<<24 * 64, 256, 0, stream>>>(
      W0, points1, points2, idx3, w3, H1, m2a);

  qbn_qrelu_kernel<<<(BATCH * COUT * NPTS) / 256, 256, 0, stream>>>(H1, m2a);

  gemm_kernel<COUT, false><<<24 * 64, 256, 0, stream>>>(
      W1, H1, nullptr, nullptr, nullptr, out, m2b);

  qbn_qrelu_kernel<<<(BATCH * COUT * NPTS) / 256, 256, 0, stream>>>(out, m2b);
}